// OneShotTopKRouter_73796128080297
// MI455X (gfx1250) — compile-verified
//
#include <hip/hip_runtime.h>
#include <hip/hip_bf16.h>

// ---------------------------------------------------------------------------
// MoE top-k router for MI455X (gfx1250, wave32).
//   logits[t,e] = sum_k hidden[t,k] * W[e,k] + b[e]     (t: 16384, e: 64, k: 2048)
//   top-8 per token -> softmax over the 8 values.
//
// Strategy: memory-bound on the 134MB hidden stream; W (512KB) stays in L2.
// Each wave computes a 16-token x 64-expert block as four 16x16 f32 C tiles
// using V_WMMA_F32_16X16X4_F32, consuming K in steps of 4.
// ---------------------------------------------------------------------------

typedef __attribute__((ext_vector_type(2))) float v2f;
typedef __attribute__((ext_vector_type(8))) float v8f;

#define HIDDEN_DIM 2048
#define N_EXPERTS  64
#define TOP_K      8
#define WAVES_PER_BLOCK 4
#define TOKENS_PER_WAVE 16
#define NEG_BIG (-3.0e38f)

__global__ __launch_bounds__(WAVES_PER_BLOCK * 32)
void OneShotTopKRouter_73796128080297_kernel(
    const float* __restrict__ hidden,   // [16384, 2048]
    const float* __restrict__ W,        // [64, 2048]
    const float* __restrict__ b,        // [64]
    float* __restrict__ out_idx,        // [16384, 8]  (indices as float)
    float* __restrict__ out_w,          // [16384, 8]
    float* __restrict__ out_logits)     // [16384, 64]
{
    __shared__ float lds_logits[WAVES_PER_BLOCK * TOKENS_PER_WAVE * N_EXPERTS];

    const int lane    = threadIdx.x & 31;
    const int wave    = threadIdx.x >> 5;
    const int tokBase = (blockIdx.x * WAVES_PER_BLOCK + wave) * TOKENS_PER_WAVE;

    const int n16   = lane & 15;          // row (A) / expert-within-tile (B,C)
    const int khalf = (lane >> 4) << 1;   // 0 for lanes 0-15, 2 for lanes 16-31

    // Per-lane base pointers. A: one token row. B: one expert row per tile.
    const float* __restrict__ Arow = hidden + (size_t)(tokBase + n16) * HIDDEN_DIM + khalf;
    const float* __restrict__ B0 = W + (size_t)( 0 + n16) * HIDDEN_DIM + khalf;
    const float* __restrict__ B1 = W + (size_t)(16 + n16) * HIDDEN_DIM + khalf;
    const float* __restrict__ B2 = W + (size_t)(32 + n16) * HIDDEN_DIM + khalf;
    const float* __restrict__ B3 = W + (size_t)(48 + n16) * HIDDEN_DIM + khalf;

    v8f acc0 = {0.f,0.f,0.f,0.f,0.f,0.f,0.f,0.f};
    v8f acc1 = acc0, acc2 = acc0, acc3 = acc0;

    #pragma unroll 4
    for (int k = 0; k < HIDDEN_DIM; k += 4) {
        // Prefetch the hidden stream ahead (global_prefetch_b8 on gfx1250).
        __builtin_prefetch(Arow + k + 512, 0, 0);

        v2f a  = *(const v2f*)(Arow + k);
        v2f b0 = *(const v2f*)(B0 + k);
        v2f b1 = *(const v2f*)(B1 + k);
        v2f b2 = *(const v2f*)(B2 + k);
        v2f b3 = *(const v2f*)(B3 + k);

        acc0 = __builtin_amdgcn_wmma_f32_16x16x4_f32(false, a, false, b0, (short)0, acc0, false, false);
        acc1 = __builtin_amdgcn_wmma_f32_16x16x4_f32(false, a, false, b1, (short)0, acc1, false, false);
        acc2 = __builtin_amdgcn_wmma_f32_16x16x4_f32(false, a, false, b2, (short)0, acc2, false, false);
        acc3 = __builtin_amdgcn_wmma_f32_16x16x4_f32(false, a, false, b3, (short)0, acc3, false, false);
    }

    // ---- Epilogue: bias add, store logits, stage tile in LDS ----
    const float bias0 = b[ 0 + n16];
    const float bias1 = b[16 + n16];
    const float bias2 = b[32 + n16];
    const float bias3 = b[48 + n16];

    // C/D layout: VGPR r, lanes 0-15 -> M=r, N=lane; lanes 16-31 -> M=r+8, N=lane-16.
    const int mBase = (lane < 16) ? 0 : 8;
    float* __restrict__ ldsW = lds_logits + wave * (TOKENS_PER_WAVE * N_EXPERTS);

    #pragma unroll
    for (int r = 0; r < 8; ++r) {
        const int m   = mBase + r;
        const int tok = tokBase + m;
        float v0 = acc0[r] + bias0;
        float v1 = acc1[r] + bias1;
        float v2 = acc2[r] + bias2;
        float v3 = acc3[r] + bias3;
        float* __restrict__ gl = out_logits + (size_t)tok * N_EXPERTS;
        gl[ 0 + n16] = v0;  gl[16 + n16] = v1;
        gl[32 + n16] = v2;  gl[48 + n16] = v3;
        float* __restrict__ ll = ldsW + m * N_EXPERTS;
        ll[ 0 + n16] = v0;  ll[16 + n16] = v1;
        ll[32 + n16] = v2;  ll[48 + n16] = v3;
    }

    __syncthreads();

    // ---- Top-8 + softmax: lanes 0-15 each own one token ----
    if (lane < 16) {
        float* __restrict__ row = ldsW + lane * N_EXPERTS;
        float vals[TOP_K];
        int   idx[TOP_K];
        #pragma unroll
        for (int i = 0; i < TOP_K; ++i) {
            float best = NEG_BIG;
            int   bi   = 0;
            for (int e = 0; e < N_EXPERTS; ++e) {
                float x = row[e];
                if (x > best) { best = x; bi = e; }   // strict '>' = first index on ties
            }
            vals[i] = best;
            idx[i]  = bi;
            row[bi] = NEG_BIG;
        }
        // softmax over the (descending) top-8; vals[0] is the max.
        const float mx = vals[0];
        float w[TOP_K];
        float s = 0.f;
        #pragma unroll
        for (int i = 0; i < TOP_K; ++i) { w[i] = __expf(vals[i] - mx); s += w[i]; }
        const float inv = 1.0f / s;

        const int tok = tokBase + lane;
        float* __restrict__ oi = out_idx + (size_t)tok * TOP_K;
        float* __restrict__ ow = out_w   + (size_t)tok * TOP_K;
        #pragma unroll
        for (int i = 0; i < TOP_K; ++i) {
            oi[i] = (float)idx[i];
            ow[i] = w[i] * inv;
        }
    }
}

extern "C" void kernel_launch(void* const* d_in, const int* in_sizes, int n_in,
                              void* d_out, int out_size, void* d_ws, size_t ws_size,
                              hipStream_t stream) {
    const float* hidden = (const float*)d_in[0];   // [4,4096,2048] -> [16384,2048]
    const float* W      = (const float*)d_in[1];   // [64,2048]
    const float* b      = (const float*)d_in[2];   // [64]

    const int n_tokens = 4 * 4096;                 // 16384
    float* out_idx    = (float*)d_out;                                   // 16384*8
    float* out_w      = out_idx + (size_t)n_tokens * TOP_K;              // 16384*8
    float* out_logits = out_w   + (size_t)n_tokens * TOP_K;              // 16384*64

    const int tokens_per_block = WAVES_PER_BLOCK * TOKENS_PER_WAVE;      // 64
    const int blocks = n_tokens / tokens_per_block;                      // 256

    OneShotTopKRouter_73796128080297_kernel<<<blocks, WAVES_PER_BLOCK * 32, 0, stream>>>(
        hidden, W, b, out_idx, out_w, out_logits);
}